// SimplePatchifier_2044404433703
// MI455X (gfx1250) — compile-verified
//
#include <hip/hip_runtime.h>
#include <stdint.h>

// Problem constants (from reference: x (64,1,512,512) f32, ps=8, k=32)
#define B_      64
#define HH      512
#define WW      512
#define PS      8
#define NP      32            // num_patches (k of top-k)
#define NH      505           // 512 - 8 + 1 sliding positions per dim
#define NCEN    (NH * NH)     // 255025 centers per batch
#define NCHUNK  32            // chunks per batch for phase 1
#define CH      7970          // ceil(NCEN / NCHUNK)
#define TPB     256
#define SEG     32            // elements owned per thread (ceil(CH/TPB))
#define SEGP    33            // padded stride -> conflict-free LDS scans
#define STAGED  (TPB * SEG)   // 8192 staged slots per chunk
#define NCAND   (NCHUNK * NP) // 1024 candidates per batch

#define NEG_INF (-__builtin_inff())

// gfx1250 async global->LDS staging (ASYNCcnt-tracked). LDS offset is the low
// 32 bits of the generic pointer (LDS aperture keeps offset in addr[31:0]).
#define USE_ASYNC_LDS 1

__global__ __launch_bounds__(TPB)
void patchify_topk_chunk(const float* __restrict__ x,
                         float* __restrict__ candVal,
                         int*   __restrict__ candIdx) {
    __shared__ float sv[TPB * SEGP];   // 8448 floats, padded chunk values
    __shared__ float rv[TPB];
    __shared__ int   rk[TPB];

    const int t = threadIdx.x;
    const int b = blockIdx.x / NCHUNK;
    const int c = blockIdx.x % NCHUNK;
    const int chunkStart = c * CH;
    const float* xb = x + (size_t)b * (HH * WW);

    // ---- Stage chunk centers into LDS (slot = k + k/32 for stride-33 pad) ----
    {
        // one prefetch per thread for the stream start (global_prefetch_b8)
        int k0 = t;
        int g0 = chunkStart + k0;
        if (g0 < NCEN) {
            int i0 = g0 / NH;
            int j0 = g0 - i0 * NH;
            __builtin_prefetch(xb + (i0 + PS / 2) * WW + (j0 + PS / 2), 0, 1);
        }
    }
    for (int k = t; k < STAGED; k += TPB) {
        int slot = k + (k >> 5);
        int gidx = chunkStart + k;
        bool ok = (k < CH) && (gidx < NCEN);
        if (ok) {
            int i = gidx / NH;
            int j = gidx - i * NH;
            const float* src = xb + (i + PS / 2) * WW + (j + PS / 2);
#if USE_ASYNC_LDS
            uint32_t ldsoff = (uint32_t)(size_t)(&sv[slot]);
            asm volatile("global_load_async_to_lds_b32 %0, %1, off"
                         :: "v"(ldsoff), "v"((uint64_t)(size_t)src)
                         : "memory");
#else
            sv[slot] = *src;
#endif
        } else {
            sv[slot] = NEG_INF;
        }
    }
#if USE_ASYNC_LDS
    asm volatile("s_wait_asynccnt 0x0" ::: "memory");
#endif
    __syncthreads();

    // ---- 32 rounds of extract-max (cached per-thread local max) ----
    float lmax;
    int   lk;
    {
        const int base = t * SEGP;
        lmax = sv[base];
        lk   = t * SEG;
#pragma unroll
        for (int m = 1; m < SEG; ++m) {
            float v = sv[base + m];
            if (v > lmax) { lmax = v; lk = t * SEG + m; }
        }
    }
    for (int r = 0; r < NP; ++r) {
        rv[t] = lmax;
        rk[t] = lk;
        __syncthreads();
        for (int s = TPB / 2; s > 0; s >>= 1) {
            if (t < s) {
                float v2 = rv[t + s];
                int   k2 = rk[t + s];
                // descending value; ties -> smaller index (matches lax.top_k)
                if (v2 > rv[t] || (v2 == rv[t] && (unsigned)k2 < (unsigned)rk[t])) {
                    rv[t] = v2; rk[t] = k2;
                }
            }
            __syncthreads();
        }
        float wv = rv[0];
        int   wk = rk[0];
        if (t == 0) {
            int o = (b * NCHUNK + c) * NP + r;
            candVal[o] = wv;
            candIdx[o] = chunkStart + wk;
        }
        if (t == (wk >> 5)) {                // winner's owner invalidates + rescans
            sv[t * SEGP + (wk & 31)] = NEG_INF;
            const int base = t * SEGP;
            lmax = sv[base];
            lk   = t * SEG;
#pragma unroll
            for (int m = 1; m < SEG; ++m) {
                float v = sv[base + m];
                if (v > lmax) { lmax = v; lk = t * SEG + m; }
            }
        }
        __syncthreads();
    }
}

__global__ __launch_bounds__(TPB)
void patchify_topk_final(const float* __restrict__ x,
                         const float* __restrict__ candVal,
                         const int*   __restrict__ candIdx,
                         float*       __restrict__ out) {
    __shared__ float cv[NCAND];
    __shared__ int   ci[NCAND];
    __shared__ float rv[TPB];
    __shared__ int   rq[TPB];
    __shared__ int   sel[NP];

    const int t = threadIdx.x;
    const int b = blockIdx.x;

    for (int q = t; q < NCAND; q += TPB) {
        cv[q] = candVal[b * NCAND + q];
        ci[q] = candIdx[b * NCAND + q];
    }
    __syncthreads();

    const int OW = NCAND / TPB;   // 4 candidates owned per thread
    float lmax;
    int   lq;
    auto rescan = [&]() {
        const int base = t * OW;
        lmax = cv[base];
        lq   = base;
#pragma unroll
        for (int m = 1; m < OW; ++m) {
            int   q = base + m;
            float v = cv[q];
            if (v > lmax || (v == lmax && ci[q] < ci[lq])) { lmax = v; lq = q; }
        }
    };
    rescan();

    for (int r = 0; r < NP; ++r) {
        rv[t] = lmax;
        rq[t] = lq;
        __syncthreads();
        for (int s = TPB / 2; s > 0; s >>= 1) {
            if (t < s) {
                float v2 = rv[t + s];
                int   q2 = rq[t + s];
                if (v2 > rv[t] || (v2 == rv[t] && ci[q2] < ci[rq[t]])) {
                    rv[t] = v2; rq[t] = q2;
                }
            }
            __syncthreads();
        }
        int wq = rq[0];
        if (t == 0) sel[r] = ci[wq];
        if (t == (wq / OW)) {
            cv[wq] = NEG_INF;
            rescan();
        }
        __syncthreads();
    }

    // ---- Gather the 32 selected 8x8 patches (x is L2-resident, 192MB L2) ----
    const float* xb = x + (size_t)b * (HH * WW);
    for (int e = t; e < NP * PS * PS; e += TPB) {
        int p  = e >> 6;          // patch rank
        int o  = e & 63;          // element within patch
        int di = o >> 3;
        int dj = o & 7;
        int g   = sel[p];
        int row = g / NH;
        int col = g - row * NH;
        out[(size_t)(b * NP + p) * (PS * PS) + o] =
            xb[(row + di) * WW + (col + dj)];
    }
}

extern "C" void kernel_launch(void* const* d_in, const int* in_sizes, int n_in,
                              void* d_out, int out_size, void* d_ws, size_t ws_size,
                              hipStream_t stream) {
    const float* x = (const float*)d_in[0];   // (64,1,512,512) f32
    // d_in[1]=patch_size(8), d_in[2]=num_patches(32): compile-time constants here.

    float* candVal = (float*)d_ws;                                        // 256 KB
    int*   candIdx = (int*)((char*)d_ws + (size_t)B_ * NCAND * sizeof(float)); // +256 KB
    float* out = (float*)d_out;

    patchify_topk_chunk<<<dim3(B_ * NCHUNK), TPB, 0, stream>>>(x, candVal, candIdx);
    patchify_topk_final<<<dim3(B_), TPB, 0, stream>>>(x, candVal, candIdx, out);
}